// KVCacheWindow_38087769981038
// MI455X (gfx1250) — compile-verified
//
#include <hip/hip_runtime.h>
#include <hip/hip_bf16.h>
#include <stdint.h>

// Problem constants (all compile-time)
#define H      32
#define L      8192           // 2^13
#define D      128
#define S      16384          // 2^14
#define G      4
#define SHIFT  (S - L)        // 8192
#define ROW4   (D / 4)        // 32 float4 per row
#define N4     (H * L * ROW4) // 8,388,608 = 2^23 float4 per tensor
#define HEAD4  (L * ROW4)     // 262,144 = 2^18 float4 per head (dst side)

typedef float __attribute__((ext_vector_type(4))) f4;
// Exact match for the async builtin's parameter type per hipcc diagnostic:
// '__attribute__((__vector_size__(4 * sizeof(int)))) int __device__ *'
typedef int i4v __attribute__((vector_size(16)));
typedef __attribute__((address_space(1))) i4v* g_i4p;   // global (AS1)
typedef __attribute__((address_space(3))) i4v* l_i4p;   // LDS (AS3)

#if __has_builtin(__builtin_amdgcn_global_load_async_to_lds_b128) && \
    __has_builtin(__builtin_amdgcn_global_store_async_from_lds_b128) && \
    __has_builtin(__builtin_amdgcn_s_wait_asynccnt)
#define USE_ASYNC_LDS 1
#else
#define USE_ASYNC_LDS 0
#endif

// out[t][h][r][:] = val_t[h][ r<4 ? r : r+8192 ][:]
// Key identity (per tensor, float4 units):
//   dst index ii = head*2^18 + r*32 + col
//   src index    = ii + head*2^18 + (r>=4 ? 2^18 : 0)
//                = ii + (ii & ~0x3FFFF) + (r>=4 ? 0x40000 : 0)
__global__ __launch_bounds__(256) void kv_window_copy(
    const float* __restrict__ k_val,
    const float* __restrict__ v_val,
    float* __restrict__ out)
{
#if USE_ASYNC_LDS
    __shared__ f4 lbuf[4 * 256];   // 16 KB staging, lane-private slots
#endif
    const f4* kv4[2] = { (const f4*)k_val, (const f4*)v_val };
    f4* out4 = (f4*)out;

    const unsigned tid  = threadIdx.x;
    const unsigned base = blockIdx.x * 1024u + tid;

    unsigned dsti[4];
#pragma unroll
    for (int j = 0; j < 4; ++j) {
        const unsigned i    = base + j * 256u;       // float4 index in concatenated K|V output
        const unsigned t    = i >> 23;               // tensor select (N4 = 2^23)
        const unsigned ii   = i & (N4 - 1u);
        const unsigned r    = (ii >> 5) & (L - 1u);  // row within head
        const unsigned add  = (ii & ~(HEAD4 - 1u)) + ((r >= G) ? HEAD4 : 0u);
        const f4* src = kv4[t] + (size_t)(ii + add);
        dsti[j] = i;
#if USE_ASYNC_LDS
        // global -> LDS, no VGPR data movement, tracked by ASYNCcnt
        __builtin_amdgcn_global_load_async_to_lds_b128(
            (g_i4p)src, (l_i4p)&lbuf[j * 256 + tid], 0, 0);
#else
        out4[i] = *src;
#endif
    }

#if USE_ASYNC_LDS
    __builtin_amdgcn_s_wait_asynccnt(0);             // loads landed in LDS
#pragma unroll
    for (int j = 0; j < 4; ++j) {
        __builtin_amdgcn_global_store_async_from_lds_b128(
            (g_i4p)&out4[dsti[j]], (l_i4p)&lbuf[j * 256 + tid], 0, 0);
    }
    __builtin_amdgcn_s_wait_asynccnt(0);             // stores drained before wave exit
#endif
}

// pos[r] = 1e9 for global tokens, else r + SHIFT (exact in f32)
__global__ __launch_bounds__(256) void kv_window_pos(float* __restrict__ pos_out)
{
    unsigned r = blockIdx.x * 256u + threadIdx.x;
    if (r < L) {
        pos_out[r] = (r < G) ? 1.0e9f : (float)(r + SHIFT);
    }
}

extern "C" void kernel_launch(void* const* d_in, const int* in_sizes, int n_in,
                              void* d_out, int out_size, void* d_ws, size_t ws_size,
                              hipStream_t stream)
{
    // setup_inputs order: k_cache(0), v_cache(1), k_val(2), v_val(3), pos(4), input_pos(5)
    const float* k_val = (const float*)d_in[2];
    const float* v_val = (const float*)d_in[3];
    float* out = (float*)d_out;

    // 2*N4 float4 chunks total; 256 threads * 4 chunks = 1024 per block -> 16384 blocks
    kv_window_copy<<<(2 * N4) / 1024, 256, 0, stream>>>(k_val, v_val, out);
    kv_window_pos<<<L / 256, 256, 0, stream>>>(out + (size_t)2 * H * L * D);
}